// GLSTM_48241072669175
// MI455X (gfx1250) — compile-verified
//
#include <hip/hip_runtime.h>
#include <math.h>

#define N_NODES 50000
#define N_EDGES 800000
#define LAYERS  2
#define DIN     128
#define HID     128

typedef __attribute__((ext_vector_type(2))) float v2f;
typedef __attribute__((ext_vector_type(8))) float v8f;

// ---------------------------------------------------------------- zero scratch
__global__ __launch_bounds__(256) void zero_f4(float* __restrict__ p, int n) {
    int i = (blockIdx.x * 256 + threadIdx.x) * 4;
    if (i < n) *(float4*)(p + i) = make_float4(0.f, 0.f, 0.f, 0.f);
}

// ------------------------------------------------- edge-parallel segment_sum
// one thread per (edge, feature); feat row stride = ld, 128 feats gathered
__global__ __launch_bounds__(256) void scatter_add(
    const float* __restrict__ feat, int ld,
    const int* __restrict__ src, const int* __restrict__ dst,
    float* __restrict__ agg) {
    long long gid = (long long)blockIdx.x * 256 + threadIdx.x;
    int e = (int)(gid >> 7);      // /128
    int t = (int)(gid & 127);
    if (e < N_EDGES) {
        int s = src[e];
        int d = dst[e];
        atomicAdd(agg + (long long)d * DIN + t, feat[(long long)s * ld + t]);
    }
}

// -------------------------------------------- fused WMMA GEMM + LSTM gates
// block: 256 thr = 8 waves; handles one (layer, 16-node tile).
// wave w owns H columns [16w, 16w+16), all 4 gates accumulated in registers.
__global__ __launch_bounds__(256) void glstm_gemm(
    const float* __restrict__ aggx,   // [N][128]
    const float* __restrict__ aggh,   // [L][N][128]
    const float* __restrict__ state,  // [L][N][384]
    const float* __restrict__ Wx,     // [L][4][128][128]
    const float* __restrict__ bx,     // [L][4][128]
    const float* __restrict__ Wh,     // [L][4][128][128]
    const float* __restrict__ bh,     // [L][4][128]
    float* __restrict__ out)          // [L][N][384]
{
    __shared__ float lax[16 * DIN];
    __shared__ float lah[16 * DIN];

    const int NT    = N_NODES / 16;          // 3125 node tiles
    const int l     = blockIdx.x / NT;
    const int nt    = blockIdx.x % NT;
    const int node0 = nt * 16;
    const int tid   = threadIdx.x;

    // cooperatively stage A tiles (rows contiguous in both src and LDS)
    const float* gax = aggx + (long long)node0 * DIN;
    const float* gah = aggh + ((long long)l * N_NODES + node0) * DIN;
    for (int f = tid * 4; f < 16 * DIN; f += 256 * 4) {
        *(float4*)(lax + f) = *(const float4*)(gax + f);
        *(float4*)(lah + f) = *(const float4*)(gah + f);
    }
    __syncthreads();

    const int wave = tid >> 5;               // wave32
    const int lane = tid & 31;
    const int h0   = wave * 16;
    const int ar   = lane & 15;              // A row / B col / D col (N)
    const int kk   = (lane >> 4) << 1;       // K sub-offset: 0 or 2

    v8f acc[4] = {{}, {}, {}, {}};           // f, i, o, z pre-activations

    const float* WX = Wx + (long long)l * 4 * DIN * HID;
    const float* WH = Wh + (long long)l * 4 * HID * HID;

    // gx contribution: agg_x (16x128) @ Wx[l,g] (128x128)
    for (int k0 = 0; k0 < DIN; k0 += 4) {
        v2f a = *(const v2f*)(lax + ar * DIN + k0 + kk);
#pragma unroll
        for (int g = 0; g < 4; ++g) {
            const float* W = WX + (long long)g * DIN * HID + (k0 + kk) * HID + h0 + ar;
            v2f b;
            b.x = W[0];
            b.y = W[HID];
            acc[g] = __builtin_amdgcn_wmma_f32_16x16x4_f32(
                false, a, false, b, (short)0, acc[g], false, false);
        }
    }
    // gh contribution: agg_h (16x128) @ Wh[l,g] (128x128), same accumulators
    for (int k0 = 0; k0 < HID; k0 += 4) {
        v2f a = *(const v2f*)(lah + ar * DIN + k0 + kk);
#pragma unroll
        for (int g = 0; g < 4; ++g) {
            const float* W = WH + (long long)g * HID * HID + (k0 + kk) * HID + h0 + ar;
            v2f b;
            b.x = W[0];
            b.y = W[HID];
            acc[g] = __builtin_amdgcn_wmma_f32_16x16x4_f32(
                false, a, false, b, (short)0, acc[g], false, false);
        }
    }

    // biases (broadcast along M); per-lane column h
    const int hcol = h0 + ar;
    const long long bb = (long long)l * 4 * HID + hcol;
    const float bsf = bx[bb + 0 * HID] + bh[bb + 0 * HID];
    const float bsi = bx[bb + 1 * HID] + bh[bb + 1 * HID];
    const float bso = bx[bb + 2 * HID] + bh[bb + 2 * HID];
    const float bsz = bx[bb + 3 * HID] + bh[bb + 3 * HID];

    const int mofs = (lane >> 4) * 8;        // C/D: M = v + 8*(lane>=16)
#pragma unroll
    for (int v = 0; v < 8; ++v) {
        const int node = node0 + v + mofs;
        const long long so = ((long long)l * N_NODES + node) * (3 * HID);
        const float c  = state[so + HID + hcol];

        const float ft = 1.f / (1.f + __expf(-(acc[0][v] + bsf)));
        const float it = 1.f / (1.f + __expf(-(acc[1][v] + bsi)));
        const float ot = 1.f / (1.f + __expf(-(acc[2][v] + bso)));
        const float zt = tanhf(acc[3][v] + bsz);

        const float ct = ft * c + it * zt;
        const float ht = ot * ct;

        out[so + hcol]           = ht;
        out[so + HID + hcol]     = ct;
        out[so + 2 * HID + hcol] = ft;
    }
}

// ---------------------------------------------------------------------------
extern "C" void kernel_launch(void* const* d_in, const int* in_sizes, int n_in,
                              void* d_out, int out_size, void* d_ws, size_t ws_size,
                              hipStream_t stream) {
    (void)in_sizes; (void)n_in; (void)out_size; (void)ws_size;
    const float* x     = (const float*)d_in[0];
    const int*   ei    = (const int*)  d_in[1];   // [2][E]
    const float* state = (const float*)d_in[2];   // [L][N][384]
    const int*   pe    = (const int*)  d_in[3];   // [2][E]
    const float* Wx    = (const float*)d_in[4];
    const float* bx    = (const float*)d_in[5];
    const float* Wh    = (const float*)d_in[6];
    const float* bh    = (const float*)d_in[7];
    float*       out   = (float*)d_out;

    float* aggx = (float*)d_ws;                              // [N][128]
    float* aggh = aggx + (size_t)N_NODES * DIN;              // [L][N][128]

    const int nz = (LAYERS + 1) * N_NODES * DIN;             // 19.2M floats
    zero_f4<<<(nz / 4 + 255) / 256, 256, 0, stream>>>(aggx, nz);

    const int sgrid = (int)(((long long)N_EDGES * 128) / 256);   // 400000
    // agg_x over edge_index
    scatter_add<<<sgrid, 256, 0, stream>>>(x, DIN, ei, ei + N_EDGES, aggx);
    // agg_h per layer over prev_edge (h = state[l][:, :128], row stride 384)
    scatter_add<<<sgrid, 256, 0, stream>>>(state, 3 * HID,
                                           pe, pe + N_EDGES, aggh);
    scatter_add<<<sgrid, 256, 0, stream>>>(state + (size_t)N_NODES * 3 * HID, 3 * HID,
                                           pe, pe + N_EDGES, aggh + (size_t)N_NODES * DIN);

    glstm_gemm<<<LAYERS * (N_NODES / 16), 256, 0, stream>>>(
        aggx, aggh, state, Wx, bx, Wh, bh, out);
}